// BackboneBuilder_40819369181394
// MI455X (gfx1250) — compile-verified
//
#include <hip/hip_runtime.h>
#include <math.h>

#define TPB    256
#define LRES   2048
#define NBATCH 256
#define NSTEP  (3*(LRES+1))            // 6147 chain-extension steps
#define SPT    ((NSTEP + TPB - 1)/TPB) // 25 steps per thread

// Per-step-phase constants (p = k%3). ac = pi - bond_angle; c=cos(ac), s=sin(ac).
// p=0: dih=psi,   angle=117.2deg (CA_C_N), len=1.336 (C_N)
// p=1: dih=omega, angle=121.7deg (C_N_CA), len=1.459 (N_CA)
// p=2: dih=phi,   angle=111.0deg (N_CA_C), len=1.525 (CA_C)
#define C0 0.4570979f
#define C1 0.5254717f
#define C2 0.3583679f
#define S0 0.8894172f
#define S1 0.8508111f
#define S2 0.9335804f
#define LB0 1.336f
#define LB1 1.459f
#define LB2 1.525f
#define OM_CD (-0.99992537f)   // cos(179.3 deg)
#define OM_SD ( 0.01221700f)   // sin(179.3 deg)
#define L_C_O 1.229f

#if defined(__AMDGCN__)
#ifdef __has_builtin
#if __has_builtin(__builtin_amdgcn_global_load_async_to_lds_b128)
#define HAVE_ASYNC_LDS 1
#endif
#endif
#endif

#ifdef HAVE_ASYNC_LDS
typedef int v4i __attribute__((ext_vector_type(4)));
typedef __attribute__((address_space(1))) v4i as1_v4i;
typedef __attribute__((address_space(3))) v4i as3_v4i;
#endif

// Compose running affine (m,t) with one extension step whose local transform is
//   M = [[c,-s,0],[s*cd,c*cd,-sd],[s*sd,c*sd,cd]],  t_step = l * M.col0
// (m,t) <- (m*M, t + l*(m*M).col0)
__device__ __forceinline__ void step_compose(float m[9], float t[3],
                                             float c, float s, float cd, float sd, float l) {
#pragma unroll
  for (int r = 0; r < 3; ++r) {
    float f0 = m[3*r+0], f1 = m[3*r+1], f2 = m[3*r+2];
    float a  = fmaf(f1, cd, f2 * sd);
    float n0 = fmaf(f0, c,  s * a);
    float n1 = fmaf(a,  c, -f0 * s);
    float n2 = fmaf(f2, cd, -f1 * sd);
    m[3*r+0] = n0; m[3*r+1] = n1; m[3*r+2] = n2;
    t[r] = fmaf(l, n0, t[r]);
  }
}

__device__ __forceinline__ void step_params(int k, const float* s_phi, const float* s_psi,
                                            float& c, float& s, float& cd, float& sd, float& l) {
  int kk = (k < 3*LRES) ? k : (k - 3);   // reference repeats the last 3 steps
  int r  = kk / 3;
  int p  = kk - 3*r;
  c = (p == 0) ? C0  : ((p == 1) ? C1  : C2);
  s = (p == 0) ? S0  : ((p == 1) ? S1  : S2);
  l = (p == 0) ? LB0 : ((p == 1) ? LB1 : LB2);
  if (p == 1) { cd = OM_CD; sd = OM_SD; }
  else {
    float dih = (p == 0) ? s_psi[r] : s_phi[r];
    __sincosf(dih, &sd, &cd);
  }
}

#define AGG_STRIDE 13  // 12 floats + 1 pad -> coprime with 64 LDS banks

__device__ __forceinline__ void aff_store(float* base, int i, const float m[9], const float t[3]) {
  float* p = base + i * AGG_STRIDE;
#pragma unroll
  for (int j = 0; j < 9; ++j) p[j] = m[j];
#pragma unroll
  for (int j = 0; j < 3; ++j) p[9 + j] = t[j];
}

__device__ __forceinline__ void aff_load(const float* base, int i, float m[9], float t[3]) {
  const float* p = base + i * AGG_STRIDE;
#pragma unroll
  for (int j = 0; j < 9; ++j) m[j] = p[j];
#pragma unroll
  for (int j = 0; j < 3; ++j) t[j] = p[9 + j];
}

// Kernel A: one block per chain. Parallel affine prefix-scan of the NeRF recurrence.
__global__ __launch_bounds__(TPB) void bb_scan_kernel(const float* __restrict__ phi,
                                                      const float* __restrict__ psi,
                                                      float* __restrict__ out,
                                                      float* __restrict__ wsN) {
  __shared__ float s_phi[LRES];               // 8 KB
  __shared__ float s_psi[LRES];               // 8 KB
  __shared__ float s_agg[TPB * AGG_STRIDE];   // 13.3 KB

  const int b   = blockIdx.x;
  const int tid = threadIdx.x;
  const float* gp = phi + (size_t)b * LRES;
  const float* gq = psi + (size_t)b * LRES;

  // ---- Stage dihedrals to LDS (CDNA5 async global->LDS DMA when available) ----
#ifdef HAVE_ASYNC_LDS
#pragma unroll
  for (int kch = 0; kch < 2; ++kch) {
    int c4 = (tid + TPB * kch) * 4;  // 16B chunk per lane
    __builtin_amdgcn_global_load_async_to_lds_b128(
        (as1_v4i*)(gp + c4), (as3_v4i*)(&s_phi[c4]), 0, 0);
    __builtin_amdgcn_global_load_async_to_lds_b128(
        (as1_v4i*)(gq + c4), (as3_v4i*)(&s_psi[c4]), 0, 0);
  }
#if __has_builtin(__builtin_amdgcn_s_wait_asynccnt)
  __builtin_amdgcn_s_wait_asynccnt(0);
#else
  asm volatile("s_wait_asynccnt 0" ::: "memory");
#endif
#else
  for (int i = tid; i < LRES; i += TPB) { s_phi[i] = gp[i]; s_psi[i] = gq[i]; }
#endif
  __syncthreads();

  // ---- Pass 1: per-thread sequential composition of its step segment ----
  float m[9] = {1.f,0.f,0.f, 0.f,1.f,0.f, 0.f,0.f,1.f};
  float t[3] = {0.f, 0.f, 0.f};
  const int k0 = tid * SPT;
  int k1 = k0 + SPT; if (k1 > NSTEP) k1 = NSTEP;
  for (int k = k0; k < k1; ++k) {
    float c, s, cd, sd, l;
    step_params(k, s_phi, s_psi, c, s, cd, sd, l);
    step_compose(m, t, c, s, cd, sd, l);
  }

  // ---- Hillis-Steele inclusive scan over affine aggregates in LDS ----
  aff_store(s_agg, tid, m, t);
  __syncthreads();
  for (int off = 1; off < TPB; off <<= 1) {
    float om[9], ot[3];
    const bool has = (tid >= off);
    if (has) aff_load(s_agg, tid - off, om, ot);
    __syncthreads();
    if (has) {
      float nm[9], nt[3];
#pragma unroll
      for (int r = 0; r < 3; ++r) {
#pragma unroll
        for (int cc = 0; cc < 3; ++cc)
          nm[3*r+cc] = fmaf(om[3*r+0], m[cc],
                       fmaf(om[3*r+1], m[3+cc], om[3*r+2] * m[6+cc]));
        nt[r] = ot[r] + fmaf(om[3*r+0], t[0],
                        fmaf(om[3*r+1], t[1], om[3*r+2] * t[2]));
      }
#pragma unroll
      for (int j = 0; j < 9; ++j) m[j] = nm[j];
#pragma unroll
      for (int j = 0; j < 3; ++j) t[j] = nt[j];
      aff_store(s_agg, tid, m, t);
    }
    __syncthreads();
  }

  // ---- Exclusive prefix, pre-composed with the initial frame F0 ----
  // F0 rows: (0,-1,0),(1,0,0),(0,0,1); x0 = 0.
  float pm[9] = {1.f,0.f,0.f, 0.f,1.f,0.f, 0.f,0.f,1.f};
  float pt[3] = {0.f, 0.f, 0.f};
  if (tid > 0) aff_load(s_agg, tid - 1, pm, pt);
  float cm[9] = { -pm[3], -pm[4], -pm[5],
                   pm[0],  pm[1],  pm[2],
                   pm[6],  pm[7],  pm[8] };
  float ct[3] = { -pt[1], pt[0], pt[2] };

  // ---- Pass 2: replay segment with global prefix, emit atom positions ----
  for (int k = k0; k < k1; ++k) {
    float c, s, cd, sd, l;
    step_params(k, s_phi, s_psi, c, s, cd, sd, l);
    step_compose(cm, ct, c, s, cd, sd, l);
    const int j = k / 3;
    const int a = k - 3 * j;
    if (j < LRES) {
      float* o = out + (((size_t)b * LRES + j) * 4 + a) * 3;
      o[0] = ct[0]; o[1] = ct[1]; o[2] = ct[2];
    } else if (a == 0) {  // N atom of residue L: needed for last O placement
      wsN[b * 3 + 0] = ct[0]; wsN[b * 3 + 1] = ct[1]; wsN[b * 3 + 2] = ct[2];
    }
  }
}

__device__ __forceinline__ void norm3(float& x, float& y, float& z) {
  float n   = sqrtf(fmaf(x, x, fmaf(y, y, z * z)));
  float inv = 1.0f / fmaxf(n, 1e-12f);
  x *= inv; y *= inv; z *= inv;
}

// Kernel B: place O atoms, accumulate per-batch coordinate sum (one block per chain).
__global__ __launch_bounds__(TPB) void bb_oxy_kernel(float* __restrict__ out,
                                                     const float* __restrict__ wsN,
                                                     float* __restrict__ wsum) {
  __shared__ float red[TPB];
  const int b   = blockIdx.x;
  const int tid = threadIdx.x;
  float acc = 0.f;
  for (int j = tid; j < LRES; j += TPB) {
    float* base = out + (((size_t)b * LRES + j) * 4) * 3;
    float Nx = base[0], Ny = base[1], Nz = base[2];
    float Ax = base[3], Ay = base[4], Az = base[5];   // CA
    float Cx = base[6], Cy = base[7], Cz = base[8];
    float nNx, nNy, nNz;
    if (j + 1 < LRES) { const float* nb = base + 12; nNx = nb[0]; nNy = nb[1]; nNz = nb[2]; }
    else              { nNx = wsN[b*3+0]; nNy = wsN[b*3+1]; nNz = wsN[b*3+2]; }
    float uax = Cx - Ax,  uay = Cy - Ay,  uaz = Cz - Az;   norm3(uax, uay, uaz);
    float ubx = Cx - nNx, uby = Cy - nNy, ubz = Cz - nNz;  norm3(ubx, uby, ubz);
    float ox = uax + ubx, oy = uay + uby, oz = uaz + ubz;  norm3(ox, oy, oz);
    float Ox = fmaf(L_C_O, ox, Cx), Oy = fmaf(L_C_O, oy, Cy), Oz = fmaf(L_C_O, oz, Cz);
    base[9] = Ox; base[10] = Oy; base[11] = Oz;
    acc += (Nx + Ny + Nz) + (Ax + Ay + Az) + (Cx + Cy + Cz) + (Ox + Oy + Oz);
  }
  red[tid] = acc;
  __syncthreads();
  for (int s = TPB / 2; s > 0; s >>= 1) {
    if (tid < s) red[tid] += red[tid + s];
    __syncthreads();
  }
  if (tid == 0) wsum[b] = red[0];
}

// Kernel C: subtract per-batch mean (mean over L*4*3 coords, matching axis=(1,2,3)).
__global__ __launch_bounds__(TPB) void bb_center_kernel(float* __restrict__ out,
                                                        const float* __restrict__ wsum) {
  size_t idx   = (size_t)blockIdx.x * TPB + threadIdx.x;
  size_t total = (size_t)NBATCH * LRES * 12;
  if (idx < total) {
    int b = (int)(idx / ((size_t)LRES * 12));
    out[idx] -= wsum[b] * (1.0f / (LRES * 12));
  }
}

extern "C" void kernel_launch(void* const* d_in, const int* in_sizes, int n_in,
                              void* d_out, int out_size, void* d_ws, size_t ws_size,
                              hipStream_t stream) {
  const float* phi = (const float*)d_in[0];
  const float* psi = (const float*)d_in[1];
  float* out  = (float*)d_out;
  float* wsN  = (float*)d_ws;          // NBATCH*3 floats: N atom of residue L
  float* wsum = wsN + NBATCH * 3;      // NBATCH floats: per-batch coordinate sums

  bb_scan_kernel<<<NBATCH, TPB, 0, stream>>>(phi, psi, out, wsN);
  bb_oxy_kernel<<<NBATCH, TPB, 0, stream>>>(out, wsN, wsum);
  size_t total = (size_t)NBATCH * LRES * 12;
  bb_center_kernel<<<(unsigned)((total + TPB - 1) / TPB), TPB, 0, stream>>>(out, wsum);
}